// SINDyLibraryOrder1_11751030522659
// MI455X (gfx1250) — compile-verified
//
#include <hip/hip_runtime.h>
#include <stdint.h>

#define D              32
#define NPAIRS         528          // C(33,2)
#define NTRIPS         5984         // C(34,3)
#define NPOLY          6545         // 1 + 32 + 528 + 5984
#define NCOLS          6577         // + 32 sine columns
#define NPAD           6560         // 205 * 32 (table padded to whole chunks)
#define NCHUNKS        205
#define ROWS_PER_BLOCK 32
#define THREADS        256
#define ZSTRIDE        36           // floats per staged row: 16B-aligned rows, slot [32] = 1.0f

// ---------------------------------------------------------------------------
// Deterministic per-call table build: packed (i,j,k) per poly column, unused
// indices = 32 (points at the 1.0f slot) so every column is z[i]*z[j]*z[k].
// ---------------------------------------------------------------------------
__global__ void build_table_kernel(unsigned int* __restrict__ tbl) {
  int c = blockIdx.x * blockDim.x + threadIdx.x;
  if (c >= NPAD) return;
  unsigned i = 32u, j = 32u, k = 32u;
  if (c == 0) {
    // constant column: 1*1*1
  } else if (c <= D) {
    i = (unsigned)(c - 1);                      // linear term
  } else if (c <= D + NPAIRS) {
    int r = c - (D + 1);                        // pair rank
    int ii = 0;
    for (;;) { int cnt = D - ii; if (r < cnt) break; r -= cnt; ++ii; }
    i = (unsigned)ii; j = (unsigned)(ii + r);
  } else if (c < NPOLY) {
    int r = c - (D + 1 + NPAIRS);               // triple rank
    int ii = 0;
    for (;;) { int cnt = (D - ii) * (D + 1 - ii) / 2; if (r < cnt) break; r -= cnt; ++ii; }
    i = (unsigned)ii;
    int jj = ii;
    for (;;) { int cnt = D - jj; if (r < cnt) break; r -= cnt; ++jj; }
    j = (unsigned)jj; k = (unsigned)(jj + r);
  }
  // columns [NPOLY, NPAD): padding (1*1*1), never stored
  tbl[c] = i | (j << 8) | (k << 16);
}

// ---------------------------------------------------------------------------
// CDNA5 async global->LDS staging (ASYNCcnt path)
// ---------------------------------------------------------------------------
__device__ __forceinline__ void async_b128(void* lds_dst, const void* gsrc) {
  unsigned l = (unsigned)(uintptr_t)lds_dst;   // low 32 bits of generic LDS ptr = LDS byte addr
  asm volatile("global_load_async_to_lds_b128 %0, %1, off"
               :: "v"(l), "v"(gsrc) : "memory");
}
__device__ __forceinline__ void wait_async0() {
  asm volatile("s_wait_asynccnt 0" ::: "memory");
}

// ---------------------------------------------------------------------------
// Main kernel: 8 wave32 per block, 32 rows per block, 4 rows per wave.
// Lane <-> output column within a 32-wide chunk => fully coalesced 128B stores.
// Output is write-once streaming -> non-temporal store hint (no L2 pollution).
// ---------------------------------------------------------------------------
__global__ __launch_bounds__(THREADS)
void sindy_lib_kernel(const float* __restrict__ z, float* __restrict__ out,
                      const unsigned int* __restrict__ tbl, int nrows) {
  __shared__ unsigned int s_tbl[NPAD];                  // 26,240 B
  __shared__ float        s_z[ROWS_PER_BLOCK][ZSTRIDE]; //  4,608 B

  const int tid  = (int)threadIdx.x;
  const int lane = tid & 31;
  const int wave = tid >> 5;
  const int row0 = (int)blockIdx.x * ROWS_PER_BLOCK;

  // Stage the block's 32 z-rows: 32x32 floats, 16B per thread, one shot.
  {
    const int r  = tid >> 3;           // 0..31
    const int c4 = (tid & 7) << 2;     // 0,4,...,28
    async_b128(&s_z[r][c4], z + (size_t)(row0 + r) * D + c4);
  }
  // Stage the index table: 6560 u32 = 1640 b128 transfers.
  for (int t = tid; t < NPAD / 4; t += THREADS) {
    async_b128(&s_tbl[t * 4], tbl + (size_t)t * 4);
  }
  if (tid < ROWS_PER_BLOCK) s_z[tid][D] = 1.0f;  // identity slot for unused indices
  wait_async0();
  __syncthreads();

  for (int rr = 0; rr < ROWS_PER_BLOCK / 8; ++rr) {
    const int lr  = wave * (ROWS_PER_BLOCK / 8) + rr;
    const int row = row0 + lr;
    if (row >= nrows) break;
    const float* zr   = s_z[lr];
    float*       orow = out + (size_t)row * NCOLS;

    // 204 full chunks: branch-free, dense 128B/wave NT stores
    #pragma unroll 4
    for (int chunk = 0; chunk < NCHUNKS - 1; ++chunk) {
      const int col = (chunk << 5) + lane;
      const unsigned e = s_tbl[col];
      const float v = zr[e & 0xffu] * zr[(e >> 8) & 0xffu] * zr[(e >> 16) & 0xffu];
      __builtin_nontemporal_store(v, &orow[col]);
    }
    // tail poly chunk (17 live lanes)
    {
      const int col = ((NCHUNKS - 1) << 5) + lane;
      if (col < NPOLY) {
        const unsigned e = s_tbl[col];
        const float v = zr[e & 0xffu] * zr[(e >> 8) & 0xffu] * zr[(e >> 16) & 0xffu];
        __builtin_nontemporal_store(v, &orow[col]);
      }
    }
    // sine chunk: columns [6545, 6577) — accurate sin to match jnp.sin
    __builtin_nontemporal_store(sinf(zr[lane]), &orow[NPOLY + lane]);
  }
}

// ---------------------------------------------------------------------------
extern "C" void kernel_launch(void* const* d_in, const int* in_sizes, int n_in,
                              void* d_out, int out_size, void* d_ws, size_t ws_size,
                              hipStream_t stream) {
  const float*  zin = (const float*)d_in[0];
  float*        out = (float*)d_out;
  unsigned int* tbl = (unsigned int*)d_ws;     // 26,240 bytes of scratch used
  const int nrows = in_sizes[0] / D;           // 65536

  build_table_kernel<<<(NPAD + THREADS - 1) / THREADS, THREADS, 0, stream>>>(tbl);

  const int nblocks = (nrows + ROWS_PER_BLOCK - 1) / ROWS_PER_BLOCK;
  sindy_lib_kernel<<<nblocks, THREADS, 0, stream>>>(zin, out, tbl, nrows);
}